// MultiHeadAttention_17188459118820
// MI455X (gfx1250) — compile-verified
//
#include <hip/hip_runtime.h>
#include <math.h>

// MI455X / gfx1250 multi-head attention, f32 end-to-end using V_WMMA_F32_16X16X4_F32.
//
// Shapes: B=2, S=2048, D=1024, H=16, Dh=64. Tokens T = B*S = 4096.
// Pipeline: (1) QKV GEMM -> scatter to Q/K/V [B*H, S, Dh] (Q pre-scaled 1/8)
//           (2) flash-style attention (online softmax), out -> [T, D]
//           (3) output GEMM + bias -> d_out.
//
// WMMA f32 16x16x4 fragment layouts (ISA 7.12.2, 32-bit):
//   A (16x4): lanes 0-15 row M=lane : VGPR0=K(2*half), VGPR1=K(2*half+1)  (half = lane>>4)
//   B (4x16): lane n = col N       : VGPR0=row(2*half), VGPR1=row(2*half+1)
//   C/D     : VGPR v -> row (v + 8*half), col (lane & 15)

typedef float v2f __attribute__((ext_vector_type(2)));
typedef float v8f __attribute__((ext_vector_type(8)));

#define HIDDEN   1024
#define HEADS    16
#define HEAD_DIM 64
#define SEQ      2048
#define BATCH    2
#define TOKENS   (BATCH * SEQ)          // 4096
#define BH       (BATCH * HEADS)        // 32
#define QKV_ONE  (BH * SEQ * HEAD_DIM)  // 4194304 floats per Q/K/V tensor

static __device__ __forceinline__ v8f wmma4(v2f a, v2f b, v8f c) {
  // 8-arg pattern: (neg_a, A, neg_b, B, c_mod, C, reuse_a, reuse_b)
  return __builtin_amdgcn_wmma_f32_16x16x4_f32(false, a, false, b, (short)0, c,
                                               false, false);
}

static __device__ __forceinline__ float redmax16(float x) {
  x = fmaxf(x, __shfl_xor(x, 1, 16));
  x = fmaxf(x, __shfl_xor(x, 2, 16));
  x = fmaxf(x, __shfl_xor(x, 4, 16));
  x = fmaxf(x, __shfl_xor(x, 8, 16));
  return x;
}
static __device__ __forceinline__ float redsum16(float x) {
  x += __shfl_xor(x, 1, 16);
  x += __shfl_xor(x, 2, 16);
  x += __shfl_xor(x, 4, 16);
  x += __shfl_xor(x, 8, 16);
  return x;
}

// ---------------------------------------------------------------------------
// Kernel 1: qkv = x @ W_qkv + b_qkv, scattered to Q/K/V [BH, S, Dh], Q *= 0.125
// Each wave computes a 16x64 output panel. 12288 waves = 3072 blocks x 128 thr.
// ---------------------------------------------------------------------------
__global__ void __launch_bounds__(128)
qkv_proj_kernel(const float* __restrict__ x, const float* __restrict__ W,
                const float* __restrict__ bias, float* __restrict__ qkv) {
  const int wid  = (blockIdx.x * blockDim.x + threadIdx.x) >> 5;
  const int lane = threadIdx.x & 31;
  const int half = lane >> 4;
  const int l16  = lane & 15;

  const int NT64 = (3 * HIDDEN) / 64;             // 48
  const int tm   = wid / NT64;                    // 0..255
  const int tn   = wid % NT64;                    // 0..47
  const int row0 = tm * 16;
  const int cb   = tn * 64;

  v8f acc[4] = {{0,0,0,0,0,0,0,0}, {0,0,0,0,0,0,0,0},
                {0,0,0,0,0,0,0,0}, {0,0,0,0,0,0,0,0}};

  const float* arow = x + (size_t)(row0 + l16) * HIDDEN + 2 * half;
#pragma unroll 2
  for (int k0 = 0; k0 < HIDDEN; k0 += 4) {
    v2f a;
    a[0] = arow[k0];
    a[1] = arow[k0 + 1];
    const float* bp = W + (size_t)(k0 + 2 * half) * (3 * HIDDEN) + cb + l16;
#pragma unroll
    for (int nt = 0; nt < 4; ++nt) {
      v2f b;
      b[0] = bp[nt * 16];
      b[1] = bp[nt * 16 + 3 * HIDDEN];
      acc[nt] = wmma4(a, b, acc[nt]);
    }
  }

#pragma unroll
  for (int nt = 0; nt < 4; ++nt) {
    const int c0    = cb + nt * 16;
    const int which = c0 >> 10;          // 0=Q 1=K 2=V
    const int rem   = c0 & 1023;
    const int h     = rem >> 6;
    const int d0    = rem & 63;
    const float sc  = (which == 0) ? 0.125f : 1.0f;
    const float bb  = bias[c0 + l16];
#pragma unroll
    for (int v = 0; v < 8; ++v) {
      const int t = row0 + v + 8 * half;           // token id
      const int bidx = t >> 11;
      const int s    = t & (SEQ - 1);
      const size_t off = (size_t)which * QKV_ONE +
                         (((size_t)(bidx * HEADS + h)) * SEQ + s) * HEAD_DIM +
                         d0 + l16;
      qkv[off] = (acc[nt][v] + bb) * sc;
    }
  }
}

// ---------------------------------------------------------------------------
// Kernel 2: flash attention. Block = 4 waves = 64 query rows of one (b,h).
// Grid = BH * (SEQ/64) = 1024 blocks. K/V 16-key tiles staged through LDS.
// ---------------------------------------------------------------------------
__global__ void __launch_bounds__(128)
attn_kernel(const float* __restrict__ Q, const float* __restrict__ K,
            const float* __restrict__ V, float* __restrict__ out) {
  __shared__ float Kt[16][68];      // pad 68: conflict-free transposed reads
  __shared__ float Vt[16][68];
  __shared__ float Ps[4][16][20];   // per-wave P staging, pad 20

  const int bh   = blockIdx.x >> 5;             // 0..31
  const int qg   = blockIdx.x & 31;             // 0..31
  const int warp = threadIdx.x >> 5;
  const int lane = threadIdx.x & 31;
  const int half = lane >> 4;
  const int l16  = lane & 15;
  const int q0   = qg * 64 + warp * 16;         // query row base within seq

  const float* Qb = Q + (size_t)bh * SEQ * HEAD_DIM;
  const float* Kb = K + (size_t)bh * SEQ * HEAD_DIM;
  const float* Vb = V + (size_t)bh * SEQ * HEAD_DIM;

  // Resident Q fragments: 16 k-chunks of 4 along Dh=64.
  v2f qf[16];
#pragma unroll
  for (int kc = 0; kc < 16; ++kc) {
    const float* qp = Qb + (size_t)(q0 + l16) * HEAD_DIM + kc * 4 + 2 * half;
    qf[kc][0] = qp[0];
    qf[kc][1] = qp[1];
  }

  v8f o[4] = {{0,0,0,0,0,0,0,0}, {0,0,0,0,0,0,0,0},
              {0,0,0,0,0,0,0,0}, {0,0,0,0,0,0,0,0}};
  float mrow[8], lrow[8];
#pragma unroll
  for (int v = 0; v < 8; ++v) { mrow[v] = -INFINITY; lrow[v] = 0.0f; }

  const int rr = threadIdx.x >> 3;           // 0..15
  const int cc = (threadIdx.x & 7) * 8;      // 0..56

  for (int kt = 0; kt < SEQ / 16; ++kt) {
    __syncthreads();
    // Cooperative, fully coalesced K/V tile load (8 floats/thread each).
    const float* kp = Kb + ((size_t)kt * 16 + rr) * HEAD_DIM + cc;
    const float* vp = Vb + ((size_t)kt * 16 + rr) * HEAD_DIM + cc;
#pragma unroll
    for (int j = 0; j < 8; ++j) {
      Kt[rr][cc + j] = kp[j];
      Vt[rr][cc + j] = vp[j];
    }
    __syncthreads();

    // S_tile = Qs @ K^T  (16x16), K-dim = Dh = 64.
    v8f st = {0,0,0,0,0,0,0,0};
#pragma unroll
    for (int kc = 0; kc < 16; ++kc) {
      v2f b;
      b[0] = Kt[l16][kc * 4 + 2 * half];
      b[1] = Kt[l16][kc * 4 + 2 * half + 1];
      st = wmma4(qf[kc], b, st);
    }

    // Online softmax update (per row v + 8*half).
#pragma unroll
    for (int v = 0; v < 8; ++v) {
      const float tmax  = redmax16(st[v]);
      const float mnew  = fmaxf(mrow[v], tmax);
      const float alpha = expf(mrow[v] - mnew);
      mrow[v] = mnew;
      const float p  = expf(st[v] - mnew);
      const float rs = redsum16(p);
      lrow[v] = lrow[v] * alpha + rs;
#pragma unroll
      for (int nt = 0; nt < 4; ++nt) o[nt][v] *= alpha;
      Ps[warp][v + 8 * half][l16] = p;      // C-layout -> LDS (same-wave)
    }

    // Re-read P in A-layout; O += P @ V_tile (K-dim = 16 keys).
    v2f pf[4];
#pragma unroll
    for (int kc = 0; kc < 4; ++kc) {
      pf[kc][0] = Ps[warp][l16][kc * 4 + 2 * half];
      pf[kc][1] = Ps[warp][l16][kc * 4 + 2 * half + 1];
    }
#pragma unroll
    for (int nt = 0; nt < 4; ++nt) {
#pragma unroll
      for (int kc = 0; kc < 4; ++kc) {
        v2f b;
        b[0] = Vt[kc * 4 + 2 * half][nt * 16 + l16];
        b[1] = Vt[kc * 4 + 2 * half + 1][nt * 16 + l16];
        o[nt] = wmma4(pf[kc], b, o[nt]);
      }
    }
  }

  // Normalize and store to [T, D] with col = h*64 + d.
  const int hh = bh & (HEADS - 1);
  const int bb = bh >> 4;
#pragma unroll
  for (int v = 0; v < 8; ++v) {
    const float inv = 1.0f / lrow[v];
    const int tkn = bb * SEQ + q0 + v + 8 * half;
#pragma unroll
    for (int nt = 0; nt < 4; ++nt) {
      out[(size_t)tkn * HIDDEN + hh * HEAD_DIM + nt * 16 + l16] = o[nt][v] * inv;
    }
  }
}

// ---------------------------------------------------------------------------
// Kernel 3: d_out = attn @ W_out + b_out.  [4096,1024] @ [1024,1024].
// ---------------------------------------------------------------------------
__global__ void __launch_bounds__(128)
out_proj_kernel(const float* __restrict__ A, const float* __restrict__ W,
                const float* __restrict__ bias, float* __restrict__ out) {
  const int wid  = (blockIdx.x * blockDim.x + threadIdx.x) >> 5;
  const int lane = threadIdx.x & 31;
  const int half = lane >> 4;
  const int l16  = lane & 15;

  const int NT64 = HIDDEN / 64;                   // 16
  const int tm   = wid / NT64;
  const int tn   = wid % NT64;
  const int row0 = tm * 16;
  const int cb   = tn * 64;

  v8f acc[4] = {{0,0,0,0,0,0,0,0}, {0,0,0,0,0,0,0,0},
                {0,0,0,0,0,0,0,0}, {0,0,0,0,0,0,0,0}};

  const float* arow = A + (size_t)(row0 + l16) * HIDDEN + 2 * half;
#pragma unroll 2
  for (int k0 = 0; k0 < HIDDEN; k0 += 4) {
    v2f a;
    a[0] = arow[k0];
    a[1] = arow[k0 + 1];
    const float* bp = W + (size_t)(k0 + 2 * half) * HIDDEN + cb + l16;
#pragma unroll
    for (int nt = 0; nt < 4; ++nt) {
      v2f b;
      b[0] = bp[nt * 16];
      b[1] = bp[nt * 16 + HIDDEN];
      acc[nt] = wmma4(a, b, acc[nt]);
    }
  }

#pragma unroll
  for (int nt = 0; nt < 4; ++nt) {
    const float bb = bias[cb + nt * 16 + l16];
#pragma unroll
    for (int v = 0; v < 8; ++v) {
      const int row = row0 + v + 8 * half;
      out[(size_t)row * HIDDEN + cb + nt * 16 + l16] = acc[nt][v] + bb;
    }
  }
}

extern "C" void kernel_launch(void* const* d_in, const int* in_sizes, int n_in,
                              void* d_out, int out_size, void* d_ws, size_t ws_size,
                              hipStream_t stream) {
  (void)in_sizes; (void)n_in; (void)out_size; (void)ws_size;
  const float* x     = (const float*)d_in[0];
  const float* W_qkv = (const float*)d_in[1];
  const float* b_qkv = (const float*)d_in[2];
  const float* W_out = (const float*)d_in[3];
  const float* b_out = (const float*)d_in[4];

  float* qkv_ws  = (float*)d_ws;                      // 3 * QKV_ONE floats (50.3 MB)
  float* attn_ws = qkv_ws + (size_t)3 * QKV_ONE;      // TOKENS*HIDDEN floats (16.8 MB)

  // K1: (TOKENS/16)*(3072/64) = 12288 waves -> 3072 blocks of 128.
  qkv_proj_kernel<<<3072, 128, 0, stream>>>(x, W_qkv, b_qkv, qkv_ws);
  // K2: BH * (SEQ/64) = 1024 blocks of 128.
  attn_kernel<<<1024, 128, 0, stream>>>(qkv_ws, qkv_ws + QKV_ONE,
                                        qkv_ws + 2 * (size_t)QKV_ONE, attn_ws);
  // K3: (TOKENS/16)*(1024/64) = 4096 waves -> 1024 blocks of 128.
  out_proj_kernel<<<1024, 128, 0, stream>>>(attn_ws, W_out, b_out, (float*)d_out);
}